// NeuralNetwork_9165460209735
// MI455X (gfx1250) — compile-verified
//
#include <hip/hip_runtime.h>
#include <cstdint>

#define WDIM 4096
#define NBLK 7
#define NCHUNK 8
#define CHUNK (WDIM / NCHUNK)        // 512 columns per wave
#define TILES (CHUNK / 4)            // 128 k-tiles of 4
#define WAVES_PER_BLOCK 8
#define ROWBLOCKS (WDIM / 16)        // 256 row-blocks of 16 rows
#define PSLOTS 10                    // 4 w-slots + 6 m-slots of 8 floats

typedef float v2f __attribute__((ext_vector_type(2)));
typedef float v8f __attribute__((ext_vector_type(8)));

// One wave owns 16 rows x 512 cols. For each 16x4 tile it issues two
// V_WMMA_F32_16X16X4_F32: A = (w*mask) resp. mask in the ISA A-layout
// (lane%16 = row, lane/16 selects K half), B column 0 = vals, column 1 = 1,
// column 2 = vals^2.  D then accumulates per-row T1/T0 (from w*m) and
// S1/n/S2 (from mask) on the matrix pipe across all 128 tiles.
__global__ __launch_bounds__(256) void masked_matvec_wmma(
    const float* __restrict__ w,
    const float* __restrict__ msk,
    const float* __restrict__ vals,
    float* __restrict__ partials)
{
    __shared__ float sv[CHUNK];

    const int tid   = threadIdx.x;
    const int wave  = tid >> 5;
    const int lane  = tid & 31;
    const int chunk = blockIdx.x;                      // 0..7
    const int rb    = blockIdx.y * WAVES_PER_BLOCK + wave; // 0..255
    const int col0  = chunk * CHUNK;

    // ---- CDNA5 async copy: stage the broadcast vals chunk into LDS ----
    if (tid < CHUNK / 4) {
        uint32_t ldso = (uint32_t)(uintptr_t)(&sv[tid * 4]);
        uint64_t ga   = (uint64_t)(uintptr_t)(vals + col0 + tid * 4);
        asm volatile("global_load_async_to_lds_b128 %0, %1, off"
                     :: "v"(ldso), "v"(ga) : "memory");
    }
    asm volatile("s_wait_asynccnt 0" ::: "memory");
    __syncthreads();

    // WMMA A-matrix 16x4 f32 layout: lanes 0-15 hold K={0,1}, lanes 16-31 K={2,3}
    const int mrow  = lane & 15;
    const int khalf = (lane >> 4) << 1;
    const size_t rowoff = (size_t)(rb * 16 + mrow) * WDIM + col0 + khalf;
    const float* wp = w   + rowoff;
    const float* mp = msk + rowoff;
    const float* vp = sv  + khalf;

    const bool n0 = (mrow == 0), n1 = (mrow == 1), n2 = (mrow == 2);

    v8f accw = {};   // D columns: 0 -> T1 = sum (w*m)*v, 1 -> T0 = sum (w*m)
    v8f accm = {};   // D columns: 0 -> S1, 1 -> n_in, 2 -> S2

#pragma unroll 8
    for (int t = 0; t < TILES; ++t) {
        v2f wv = *(const v2f*)(wp + t * 4);
        v2f mv = *(const v2f*)(mp + t * 4);
        v2f vv = *(const v2f*)(vp + t * 4);          // broadcast ds_load_b64
        v2f av = wv * mv;                            // w ⊙ mask
        float b0 = n0 ? vv.x : (n1 ? 1.0f : (n2 ? vv.x * vv.x : 0.0f));
        float b1 = n0 ? vv.y : (n1 ? 1.0f : (n2 ? vv.y * vv.y : 0.0f));
        v2f bv = { b0, b1 };
        accw = __builtin_amdgcn_wmma_f32_16x16x4_f32(
                   false, av, false, bv, (short)0, accw, false, false);
        accm = __builtin_amdgcn_wmma_f32_16x16x4_f32(
                   false, mv, false, bv, (short)0, accm, false, false);
    }

    // D layout: VGPR v, lanes 0-15 = (M=v, N=lane); lanes 16-31 = (M=v+8, N=lane-16).
    // Only N in {0,1,2} is meaningful -> lanes 0,1,2 and 16,17,18 carry results.
    float* pb = partials + ((size_t)rb * NCHUNK + chunk) * (PSLOTS * 8);
    if      (lane == 0)  { *(v8f*)(pb + 0 * 8) = accw; *(v8f*)(pb + 4 * 8) = accm; }
    else if (lane == 1)  { *(v8f*)(pb + 1 * 8) = accw; *(v8f*)(pb + 5 * 8) = accm; }
    else if (lane == 2)  {                              *(v8f*)(pb + 6 * 8) = accm; }
    else if (lane == 16) { *(v8f*)(pb + 2 * 8) = accw; *(v8f*)(pb + 7 * 8) = accm; }
    else if (lane == 17) { *(v8f*)(pb + 3 * 8) = accw; *(v8f*)(pb + 8 * 8) = accm; }
    else if (lane == 18) {                              *(v8f*)(pb + 9 * 8) = accm; }
}

// Reduce the 8 column-chunks per row and apply the fused normalization +
// affine + activation epilogue.  Also provides the grid-wide layer sync.
__global__ __launch_bounds__(256) void layer_epilogue(
    const float* __restrict__ partials,
    const float* __restrict__ gamma, const float* __restrict__ beta,
    const float* __restrict__ bias,  const float* __restrict__ gain,
    const float* __restrict__ amp,
    float* __restrict__ out, int last)
{
    int r = blockIdx.x * blockDim.x + threadIdx.x;
    if (r >= WDIM) return;
    int rb = r >> 4, j = r & 15;
    int hi = j >> 3;       // row in upper half of the 16-row block?
    int e  = j & 7;        // element (VGPR index) inside the stored v8f

    float T1 = 0.f, T0 = 0.f, S1 = 0.f, SN = 0.f, S2 = 0.f;
    for (int c = 0; c < NCHUNK; ++c) {
        const float* p = partials + ((size_t)rb * NCHUNK + c) * (PSLOTS * 8);
        T1 += p[(0 + 2 * hi) * 8 + e];
        T0 += p[(1 + 2 * hi) * 8 + e];
        S1 += p[(4 + 3 * hi) * 8 + e];
        SN += p[(5 + 3 * hi) * 8 + e];
        S2 += p[(6 + 3 * hi) * 8 + e];
    }
    float mean = S1 / SN;
    float var  = S2 / SN - mean * mean;
    float rs   = rsqrtf(var + 1e-5f);
    float affine = gamma[r] * rs * (T1 - mean * T0) + beta[r] * T0 + bias[r];
    float z = affine * gain[r];
    float o = last ? (z * amp[r])
                   : (z / (1.0f + __expf(-z))) * amp[r];   // silu(z)*amp
    out[r] = o;
}

extern "C" void kernel_launch(void* const* d_in, const int* in_sizes, int n_in,
                              void* d_out, int out_size, void* d_ws, size_t ws_size,
                              hipStream_t stream) {
    const float* x       = (const float*)d_in[0];
    const float* weights = (const float*)d_in[1];
    const float* masks   = (const float*)d_in[2];
    const float* biases  = (const float*)d_in[3];
    const float* gamma   = (const float*)d_in[4];
    const float* beta    = (const float*)d_in[5];
    const float* gain    = (const float*)d_in[6];
    const float* amp     = (const float*)d_in[7];

    float* P  = (float*)d_ws;                              // 256*8*10*8 floats
    float* vA = P + (size_t)ROWBLOCKS * NCHUNK * PSLOTS * 8;
    float* vB = vA + WDIM;

    const float* vin = x;
    for (int i = 0; i < NBLK; ++i) {
        float* vout = (i == NBLK - 1) ? (float*)d_out : ((i & 1) ? vB : vA);
        dim3 grid(NCHUNK, ROWBLOCKS / WAVES_PER_BLOCK);    // (8, 32)
        masked_matvec_wmma<<<grid, 256, 0, stream>>>(
            weights + (size_t)i * WDIM * WDIM,
            masks   + (size_t)i * WDIM * WDIM,
            vin, P);
        layer_epilogue<<<(WDIM + 255) / 256, 256, 0, stream>>>(
            P, gamma + i * WDIM, beta + i * WDIM, biases + i * WDIM,
            gain + i * WDIM, amp + i * WDIM, vout, i == NBLK - 1 ? 1 : 0);
        vin = vout;
    }
}